// Subtraction2_48009144435375
// MI455X (gfx1250) — compile-verified
//
#include <hip/hip_runtime.h>
#include <stdint.h>

// SAN subtraction2: out[n,c,kk,l] = reflectpad3(input2)[window] - input1[l]
// N=8, C=64, H=W=56, K=7, S=1, P=3, D=1  -> out [8,64,49,3136] fp32
//
// Roofline: 315 MB stores vs 12.8 MB input reads -> pure HBM-store bound
// (~13.5 us at 23.3 TB/s). Design:
//   - grid (512 planes x 7 kernel-rows); per-block LDS staging of both
//     12.5 KB planes via CDNA5 global_load_async_to_lds_b128 (ASYNCcnt)
//   - per output float4: load the 10 distinct key values covering all 7
//     column taps once from LDS, then 7 taps = 4 subs + 1 NT b128 store
//     with compile-time immediate plane offsets
//   - non-temporal b128 stores for the write-once 315 MB output stream

#define KSZ      7
#define PADW     3
#define HW       56
#define PLANE    3136           // 56*56 floats per (n,c) plane
#define PLANE4   784            // PLANE/4 float4s
#define KK       49
#define OUTPLANE (KK * PLANE)   // 153664 floats per output plane
#define NTHREADS 256

typedef float __attribute__((ext_vector_type(4))) vf4;

__device__ __forceinline__ int reflect56(int t) {
    // valid for t in [-3, 58]: mirror-without-edge (np/jax 'reflect')
    int a = t < 0 ? -t : t;          // left mirror
    return a < HW ? a : 110 - a;     // right mirror (2*56-2 - t)
}

__device__ __forceinline__ void async_f4_to_lds(const vf4* gsrc, uint32_t lds_byte_off) {
    // VDST = VGPR with LDS byte offset, VADDR = 64-bit global address (GV mode)
    asm volatile("global_load_async_to_lds_b128 %0, %1, off"
                 :: "v"(lds_byte_off), "v"(gsrc)
                 : "memory");
}

__global__ __launch_bounds__(NTHREADS)
void sub2_kernel(const float* __restrict__ in1,
                 const float* __restrict__ in2,
                 float* __restrict__ out) {
    __shared__ __align__(16) float s_key[PLANE];   // input2 plane (keys)
    __shared__ __align__(16) float s_qry[PLANE];   // input1 plane (queries)

    const int plane = blockIdx.x;     // n*C + c, 0..511
    const int ki    = blockIdx.y;     // kernel row 0..6
    const int di    = ki - PADW;      // row tap offset, block-uniform
    const int tid   = threadIdx.x;

    const float* g2 = in2 + (size_t)plane * PLANE;
    const float* g1 = in1 + (size_t)plane * PLANE;
    float*      opk = out + (size_t)plane * OUTPLANE + (size_t)ki * KSZ * PLANE;

    // Low 32 bits of a generic pointer into LDS == LDS byte offset
    const uint32_t skey_base = (uint32_t)(uintptr_t)(void*)s_key;
    const uint32_t sqry_base = (uint32_t)(uintptr_t)(void*)s_qry;

    // ---- async-stage both 12.5 KB planes into LDS (ASYNCcnt-tracked) ----
    for (int v = tid; v < PLANE4; v += NTHREADS) {
        async_f4_to_lds((const vf4*)g2 + v, skey_base + (uint32_t)v * 16u);
        async_f4_to_lds((const vf4*)g1 + v, sqry_base + (uint32_t)v * 16u);
    }
    asm volatile("s_wait_asynccnt 0" ::: "memory");
    __syncthreads();

    // ---- compute: each thread-iteration emits 7 taps x one float4 ----
    // 56 % 4 == 0 -> every float4 lies inside one output row (oh fixed).
    for (int r4 = tid; r4 < PLANE4; r4 += NTHREADS) {
        int oh = r4 / 14;                    // l/56 with l = 4*r4
        int ow = (r4 - oh * 14) << 2;        // column of first lane elem
        int l  = r4 << 2;

        const float* srow = &s_key[reflect56(oh + di) * HW];

        // The 7 column taps over 4 lanes touch exactly 10 key values:
        // reflect(ow + t - 3), t = 0..9  (register array, fully unrolled)
        float rv[KSZ + 3];
        #pragma unroll
        for (int t = 0; t < KSZ + 3; ++t)
            rv[t] = srow[reflect56(ow + t - PADW)];

        vf4 q = *(const vf4*)&s_qry[l];      // aligned ds_load_b128

        #pragma unroll
        for (int j = 0; j < KSZ; ++j) {      // column tap dj = j - 3
            vf4 r;
            r.x = rv[j + 0] - q.x;
            r.y = rv[j + 1] - q.y;
            r.z = rv[j + 2] - q.z;
            r.w = rv[j + 3] - q.w;
            // same base address, immediate offset j*12544B (fits IOFFSET)
            __builtin_nontemporal_store(r, (vf4*)(opk + (size_t)j * PLANE + l));
        }
    }
}

extern "C" void kernel_launch(void* const* d_in, const int* in_sizes, int n_in,
                              void* d_out, int out_size, void* d_ws, size_t ws_size,
                              hipStream_t stream) {
    const float* in1 = (const float*)d_in[0];  // queries [N,C,H,W]
    const float* in2 = (const float*)d_in[1];  // keys    [N,C,H,W]
    float* out = (float*)d_out;                // [N,C,49,3136]

    const int nplanes = in_sizes[0] / PLANE;   // N*C = 512
    dim3 grid(nplanes, KSZ);                   // 512 x 7 = 3584 blocks
    sub2_kernel<<<grid, NTHREADS, 0, stream>>>(in1, in2, out);
}